// ROIPooling_21775484191049
// MI455X (gfx1250) — compile-verified
//
#include <hip/hip_runtime.h>

typedef float v4f __attribute__((ext_vector_type(4)));

#define POOL 7
#define FM_H 64
#define FM_W 64
#define FM_C 128

// One block per ROI, 224 threads = 7 wave32s: wave j owns output row j,
// lane l owns channels [4l, 4l+4). Inner i-loop is fully unrolled so each
// wave issues up to 28 independent global_load_b128 gathers (512B/wave
// segments) before blending, maximizing memory-level parallelism.
__global__ __launch_bounds__(224) void roi_bilinear_kernel(
    const float* __restrict__ fm,
    const int*   __restrict__ rois,
    float*       __restrict__ out,
    int n)
{
    const int roi = blockIdx.x;
    if (roi >= n) return;

    // Block-uniform ROI fetch -> scalar s_load_b128.
    const int4 rb = reinterpret_cast<const int4*>(rois)[roi];
    const int rx1 = rb.x, ry1 = rb.y, rx2 = rb.z, ry2 = rb.w;

    const int lane = threadIdx.x & 31;
    const int j    = threadIdx.x >> 5;   // output row, one wave per row
    const int c    = lane << 2;          // 4 channels per lane

    float* outRow = out + (size_t)roi * (POOL * POOL * FM_C) + j * (POOL * FM_C);

    // Reference zeroes the output when the whole ROI is zero.
    if ((rx1 | ry1 | rx2 | ry2) == 0) {
        v4f z = {0.0f, 0.0f, 0.0f, 0.0f};
#pragma unroll
        for (int i = 0; i < POOL; ++i)
            __builtin_nontemporal_store(z, (v4f*)(outRow + i * FM_C + c));
        return;
    }

    const int hh = max(ry2 - ry1, 1);
    const int ww = max(rx2 - rx1, 1);
    const float hf = (float)hh;
    const float wf = (float)ww;

    // CDNA5 global_prefetch_b8: warm the crop's rows (<=32 rows; each
    // enabled lane prefetches one row-start cacheline).
    if (j == 0 && lane < hh) {
        __builtin_prefetch(fm + ((ry1 + lane) * FM_W + rx1) * FM_C, 0, 1);
    }

    // Row (y) math: once per wave, hoisted out of the unrolled i-loop.
    // Half-pixel centers matching tf.image.resize bilinear:
    //   yc = clip((j+0.5)*hf/POOL - 0.5, 0, hf-1)
    float yc = ((float)j + 0.5f) * hf / (float)POOL - 0.5f;
    yc = fminf(fmaxf(yc, 0.0f), hf - 1.0f);
    const int   y0 = (int)floorf(yc);
    const float wy = yc - (float)y0;
    const int   y1 = min(y0 + 1, hh - 1);
    const int   ya0 = (y0 + ry1) * FM_W;
    const int   ya1 = (y1 + ry1) * FM_W;

#pragma unroll
    for (int i = 0; i < POOL; ++i) {
        float xc = ((float)i + 0.5f) * wf / (float)POOL - 0.5f;
        xc = fminf(fmaxf(xc, 0.0f), wf - 1.0f);
        const int   x0 = (int)floorf(xc);
        const float wx = xc - (float)x0;
        const int   x1 = min(x0 + 1, ww - 1);
        const int   xa0 = x0 + rx1;
        const int   xa1 = x1 + rx1;

        // 4 corner gathers: global_load_b128, 512B coalesced per wave.
        const v4f g00 = *(const v4f*)(fm + (ya0 + xa0) * FM_C + c);
        const v4f g01 = *(const v4f*)(fm + (ya0 + xa1) * FM_C + c);
        const v4f g10 = *(const v4f*)(fm + (ya1 + xa0) * FM_C + c);
        const v4f g11 = *(const v4f*)(fm + (ya1 + xa1) * FM_C + c);

        const v4f top = g00 + wx * (g01 - g00);
        const v4f bot = g10 + wx * (g11 - g10);
        const v4f res = top + wy * (bot - top);

        // Write-once 251MB stream: non-temporal so the 2MB feature map
        // stays resident in L2/WGP$ (global_store_b128 th:TH_STORE_NT).
        __builtin_nontemporal_store(res, (v4f*)(outRow + i * FM_C + c));
    }
}

extern "C" void kernel_launch(void* const* d_in, const int* in_sizes, int n_in,
                              void* d_out, int out_size, void* d_ws, size_t ws_size,
                              hipStream_t stream) {
    const float* fm   = (const float*)d_in[0];   // (1,64,64,128) f32
    const int*   rois = (const int*)d_in[1];     // (1,N,4) i32
    float*       out  = (float*)d_out;           // (1,N,7,7,128) f32

    const int n = in_sizes[1] / 4;               // N = 10000
    roi_bilinear_kernel<<<dim3(n), dim3(224), 0, stream>>>(fm, rois, out, n);
}